// Kan1xN_34110630265540
// MI455X (gfx1250) — compile-verified
//
#include <hip/hip_runtime.h>
#include <hip/hip_bf16.h>
#include <math.h>

typedef __attribute__((ext_vector_type(2))) float v2f;
typedef __attribute__((ext_vector_type(4))) float v4f;
typedef __attribute__((ext_vector_type(8))) float v8f;

namespace {
constexpr int NPAR   = 128;     // params per output (K)
constexpr int NOUT   = 1024;    // output rows (M)
constexpr int NX     = 262144;  // x samples (N)
constexpr int OTILE  = 64;      // output rows per block  (4 wave-rows of 16)
constexpr int NTILE  = 32;      // x columns per inner tile (2 wave-cols of 16)
constexpr int NCHUNK = 1024;    // x columns per block (amortizes A-fragment load)
constexpr int LSTR   = 132;     // padded LDS row stride in floats:
                                // 132*4B = 528B -> 16B aligned, and 132%64 != 0
                                // so 16 lanes reading same k, different rows hit
                                // 16 distinct banks (no LDS bank conflicts)
}

// Clamped cubic B-spline knots: k[0..2]=0, k[3..126]=linspace(0,1,124), k[127..130]=1
__device__ __forceinline__ float knot(int i) {
  int t = i - 3;
  t = t < 0 ? 0 : (t > 123 ? 123 : t);
  return (float)t * (1.0f / 123.0f);
}

__global__ __launch_bounds__(256) void kan_wmma_f32(
    const float* __restrict__ x, const float* __restrict__ params,
    float* __restrict__ out) {
  __shared__ __align__(16) float A_lds[OTILE * LSTR];   // params tile 64 x 128 (padded)
  __shared__ __align__(16) float B_lds[NTILE * LSTR];   // dense basis tile 32 cols x 128 k (padded)

  const int tid  = threadIdx.x;
  const int lane = tid & 31;
  const int wave = tid >> 5;          // 0..7
  const int oBase = blockIdx.y * OTILE;
  const int nChunkBase = blockIdx.x * NCHUNK;

  // ---------------- stage params tile to LDS (coalesced b128) ----------------
  for (int i = tid; i < OTILE * (NPAR / 4); i += 256) {
    const int r  = i / (NPAR / 4);
    const int c4 = i % (NPAR / 4);
    v4f v = *(const v4f*)(params + (size_t)(oBase + r) * NPAR + c4 * 4);
    *(v4f*)(&A_lds[r * LSTR + c4 * 4]) = v;
  }
  __syncthreads();

  // ---------------- load this wave's A fragments into VGPRs ------------------
  // WMMA f32 16x16x4 A layout (ISA 7.12.2): lane = M (0..15 repeated),
  // VGPR0 holds K=0 (lanes 0-15) / K=2 (lanes 16-31), VGPR1 holds K=1 / K=3.
  const int mRow = (wave >> 1) * 16 + (lane & 15);  // row in 64-row tile
  const int ksel = (lane < 16) ? 0 : 2;
  v2f afrag[NPAR / 4];
  #pragma unroll
  for (int kk = 0; kk < NPAR / 4; ++kk) {
    const float* p = &A_lds[mRow * LSTR + kk * 4 + ksel];
    afrag[kk] = v2f{p[0], p[1]};
  }
  __syncthreads();

  // Per-thread basis-construction assignment: column = tid%32, k-segment = tid/32
  const int colLocal = tid & 31;      // which x column this thread evaluates
  const int kseg     = tid >> 5;      // which 16-wide k segment it fills (0..7)

  // Per-wave WMMA read/write coordinates
  const int nColLocal = (wave & 1) * 16 + (lane & 15);           // B: lane = N
  const float* bBase  = &B_lds[nColLocal * LSTR + ksel];
  const int cRow      = oBase + (wave >> 1) * 16 + ((lane < 16) ? 0 : 8);
  const int cColLocal = (wave & 1) * 16 + (lane & 15);

  for (int it = 0; it < NCHUNK / NTILE; ++it) {
    const int nBase = nChunkBase + it * NTILE;
    if (it + 1 < NCHUNK / NTILE)
      __builtin_prefetch(x + nBase + NTILE, 0, 0);   // global_prefetch_b8

    // -------- build dense basis tile: 4-nonzero cubic B-spline + SiLU row ----
    {
      const float xv = x[nBase + colLocal];
      float t = xv * 123.0f;
      int j = (int)t;                 // knot interval (interior index)
      j = j < 0 ? 0 : (j > 122 ? 122 : j);
      const int m = j + 3;

      // de Boor "BasisFuns": w[0..3] = B_{j..j+3, 3}(xv); denominators never 0
      float w[4], left[4], right[4];
      w[0] = 1.0f;
      #pragma unroll
      for (int d = 1; d <= 3; ++d) {
        left[d]  = xv - knot(m + 1 - d);
        right[d] = knot(m + d) - xv;
        float saved = 0.0f;
        #pragma unroll
        for (int r = 0; r < 3; ++r) {
          if (r < d) {
            float temp = w[r] / (right[r + 1] + left[d - r]);
            w[r] = saved + right[r + 1] * temp;
            saved = left[d - r] * temp;
          }
        }
        w[d] = saved;
      }
      const float silu = xv / (1.0f + __expf(-xv));

      float vals[16];
      #pragma unroll
      for (int i = 0; i < 16; ++i) {
        const int k = kseg * 16 + i;
        const int d = k - j;
        float v = 0.0f;
        if (d >= 0 && d < 4 && k < 127) v = w[d];
        if (k == 127) v = silu;
        vals[i] = v;
      }
      float* dst = &B_lds[colLocal * LSTR + kseg * 16];
      #pragma unroll
      for (int i = 0; i < 4; ++i)
        *(v4f*)(dst + i * 4) = v4f{vals[4 * i], vals[4 * i + 1],
                                   vals[4 * i + 2], vals[4 * i + 3]};
    }
    __syncthreads();

    // -------- 32 x v_wmma_f32_16x16x4_f32 accumulating a 16x16 fp32 tile -----
    v8f c = {0.f, 0.f, 0.f, 0.f, 0.f, 0.f, 0.f, 0.f};
    #pragma unroll
    for (int kk = 0; kk < NPAR / 4; ++kk) {
      const float* bp = bBase + kk * 4;
      v2f b = v2f{bp[0], bp[1]};     // B: VGPR0 = K=ksel row, VGPR1 = K=ksel+1
      c = __builtin_amdgcn_wmma_f32_16x16x4_f32(
              /*neg_a=*/false, afrag[kk], /*neg_b=*/false, b,
              /*c_mod=*/(short)0, c, /*reuse_a=*/false, /*reuse_b=*/false);
    }
    __syncthreads();   // all reads of B_lds done before next tile overwrites it

    // -------- store C: lane = N, VGPR v -> M = v + (lane<16 ? 0 : 8) ---------
    const size_t outBase = (size_t)cRow * NX + (size_t)(nBase + cColLocal);
    #pragma unroll
    for (int v = 0; v < 8; ++v)
      out[outBase + (size_t)v * NX] = c[v];
  }
}

extern "C" void kernel_launch(void* const* d_in, const int* in_sizes, int n_in,
                              void* d_out, int out_size, void* d_ws, size_t ws_size,
                              hipStream_t stream) {
  const float* x      = (const float*)d_in[0];   // (262144,) fp32
  const float* params = (const float*)d_in[1];   // (1024, 128) fp32
  float* out          = (float*)d_out;           // (1024, 262144) fp32

  dim3 grid(NX / NCHUNK, NOUT / OTILE);          // (256, 16)
  kan_wmma_f32<<<grid, 256, 0, stream>>>(x, params, out);

  (void)in_sizes; (void)n_in; (void)out_size; (void)d_ws; (void)ws_size;
}